// SymmetricContraction_9964324127547
// MI455X (gfx1250) — compile-verified
//
#include <hip/hip_runtime.h>

typedef __attribute__((ext_vector_type(2))) float v2f;
typedef __attribute__((ext_vector_type(8))) float v8f;

#define FEAT 128
#define NTYPE 10
#define DIM 16
#define GNODES 4   // nodes per apply-block: 4x reduction of C3 L2 traffic

// ---------------------------------------------------------------------------
// Stage 1: C[R,128] = U[R,K] @ W[e][K,128]   (fp32 WMMA, 16x16 tile per wave)
//   V_WMMA_F32_16X16X4_F32: A = 16x4 fp32 (2 VGPR), B = 4x16 fp32 (2 VGPR),
//   C/D = 16x16 fp32 (8 VGPR). K in {4,12,40} -> all multiples of 4.
// ---------------------------------------------------------------------------
__global__ void __launch_bounds__(32)
cg_weight_gemm_wmma(const float* __restrict__ U,   // [R, K] row-major
                    const float* __restrict__ W,   // [NTYPE, K, FEAT]
                    float* __restrict__ C,         // [R, FEAT]
                    int R, int K, int e) {
  const int lane = threadIdx.x & 31;
  const int half = lane >> 4;   // K pair {0,1} vs {2,3} within k-tile
  const int l16  = lane & 15;   // M index (A) / N index (B)
  const int r0 = blockIdx.x * 16;
  const int c0 = blockIdx.y * 16;

  const float* Wt = W + (size_t)e * K * FEAT;

  v8f acc = {};
  for (int k0 = 0; k0 < K; k0 += 4) {
    v2f a, b;
    const float* up = U + (size_t)(r0 + l16) * K + (k0 + 2 * half);
    a.x = up[0];
    a.y = up[1];
    const float* wp = Wt + (size_t)(k0 + 2 * half) * FEAT + (c0 + l16);
    b.x = wp[0];
    b.y = wp[FEAT];
    acc = __builtin_amdgcn_wmma_f32_16x16x4_f32(false, a, false, b, (short)0,
                                                acc, false, false);
  }
#pragma unroll
  for (int j = 0; j < 8; ++j) {
    C[(size_t)(r0 + j + 8 * half) * FEAT + (c0 + l16)] = acc[j];
  }
}

// ---------------------------------------------------------------------------
// Node bucketing: counts[t] = #nodes of type t, lists[t*B_N + i] = node ids.
// Intra-list order is atomic-arrival order (irrelevant: nodes independent).
// ---------------------------------------------------------------------------
__global__ void reset_counts(int* counts) {
  if (threadIdx.x < NTYPE) counts[threadIdx.x] = 0;
}

__global__ void build_type_lists(const float* __restrict__ node_types,
                                 int* __restrict__ counts,
                                 int* __restrict__ lists, int B_N) {
  const int b = blockIdx.x * blockDim.x + threadIdx.x;
  if (b >= B_N) return;
  int t = 0;
  for (int j = 0; j < NTYPE; ++j)
    if (node_types[b * NTYPE + j] > 0.5f) t = j;
  const int pos = atomicAdd(&counts[t], 1);
  lists[t * B_N + pos] = b;
}

// ---------------------------------------------------------------------------
// Stage 2: apply the contraction chain for GNODES nodes of type e at once.
// Thread = channel c; each coalesced 512B C3 row line feeds GNODES FMA chains:
//   t2(r2) = sum_i3 C3[r2,i3]*a_i3 + C2[r2]
//   t1(r1) = sum_i2 t2(r1,i2)*a_i2 + C1[r1]
//   out(m) = sum_i1 t1(m,i1)*a_i1
// ---------------------------------------------------------------------------
__global__ void __launch_bounds__(FEAT)
apply_contraction_grouped(const float* __restrict__ A,      // [B_N, FEAT, DIM]
                          const float* __restrict__ C3,     // [65536, FEAT]
                          const float* __restrict__ C2,     // [4096,  FEAT]
                          const float* __restrict__ C1,     // [256,   FEAT]
                          const int* __restrict__ counts,
                          const int* __restrict__ lists,
                          float* __restrict__ out,          // [B_N, FEAT, 16]
                          int e, int B_N) {
  const int c = threadIdx.x;
  const int n_e = counts[e];
  const int g0 = blockIdx.x * GNODES;
  if (g0 >= n_e) return;

  int  nb[GNODES];
  bool act[GNODES];
  float a[GNODES][DIM];
#pragma unroll
  for (int j = 0; j < GNODES; ++j) {
    act[j] = (g0 + j) < n_e;
    nb[j]  = lists[e * B_N + (act[j] ? g0 + j : g0)];
#pragma unroll
    for (int i = 0; i < DIM; ++i)
      a[j][i] = act[j] ? A[((size_t)nb[j] * FEAT + c) * DIM + i] : 0.f;
  }

  const int base3[4] = {0, 4096, 16384, 36864};  // unit 16^3 rows per m
  const int base2[4] = {0, 256, 1024, 2304};     // unit 16^2 rows per m
  const int base1[4] = {0, 16, 64, 144};         // unit 16   rows per m

  int mo = 0;
  for (int l = 0; l < 4; ++l) {
    const int mcount = (l == 0) ? 1 : (2 * l + 1);
    const float* C3l = C3 + (size_t)base3[l] * FEAT + c;
    const float* C2l = C2 + (size_t)base2[l] * FEAT + c;
    const float* C1l = C1 + (size_t)base1[l] * FEAT + c;
    for (int m = 0; m < mcount; ++m) {
      float out_m[GNODES] = {};
      for (int i1 = 0; i1 < DIM; ++i1) {
        float t1[GNODES] = {};
        for (int i2 = 0; i2 < DIM; ++i2) {
          const int r2 = (m * 16 + i1) * 16 + i2;
          const float* p = C3l + (size_t)(r2 * 16) * FEAT;
          // prefetch next row-block of C3 (global_prefetch_b8)
          __builtin_prefetch(p + (size_t)16 * FEAT, 0, 1);
          float t2[GNODES] = {};
#pragma unroll
          for (int i3 = 0; i3 < DIM; ++i3) {
            const float v = p[(size_t)i3 * FEAT];
#pragma unroll
            for (int j = 0; j < GNODES; ++j)
              t2[j] = fmaf(v, a[j][i3], t2[j]);
          }
          const float c2v = C2l[(size_t)r2 * FEAT];
#pragma unroll
          for (int j = 0; j < GNODES; ++j) {
            t2[j] += c2v;
            t1[j] = fmaf(t2[j], a[j][i2], t1[j]);
          }
        }
        const float c1v = C1l[(size_t)(m * 16 + i1) * FEAT];
#pragma unroll
        for (int j = 0; j < GNODES; ++j) {
          t1[j] += c1v;
          out_m[j] = fmaf(t1[j], a[j][i1], out_m[j]);
        }
      }
#pragma unroll
      for (int j = 0; j < GNODES; ++j)
        if (act[j])
          out[((size_t)nb[j] * FEAT + c) * DIM + (mo + m)] = out_m[j];
    }
    mo += mcount;
  }
}

// ---------------------------------------------------------------------------
// Launch: bucket nodes by type once, then per type e (stream-serialized so the
// ~36 MB C workspace is reused and C3 stays L2-resident):
//   12 WMMA GEMMs (4 irreps x 3 correlation orders) -> C1/C2/C3
//   grouped apply over that type's node list (GNODES nodes per block)
// ---------------------------------------------------------------------------
extern "C" void kernel_launch(void* const* d_in, const int* in_sizes, int n_in,
                              void* d_out, int out_size, void* d_ws,
                              size_t ws_size, hipStream_t stream) {
  const float* A  = (const float*)d_in[0];
  const float* nt = (const float*)d_in[1];
  const int B_N = in_sizes[0] / (FEAT * DIM);

  // workspace layout: C3 | C2 | C1 (floats, ~35.7 MB) | counts | lists (ints)
  float* C3 = (float*)d_ws;
  float* C2 = C3 + (size_t)65536 * FEAT;
  float* C1 = C2 + (size_t)4096 * FEAT;
  int* counts = (int*)(C1 + (size_t)256 * FEAT);
  int* lists  = counts + NTYPE;

  const int  Kn[3]        = {4, 12, 40};      // coupling paths per nu
  const int  rows_unit[3] = {16, 256, 4096};  // U rows per m for nu=1,2,3
  const int  base1[4]     = {0, 16, 64, 144};
  const int  base2[4]     = {0, 256, 1024, 2304};
  const int  base3[4]     = {0, 4096, 16384, 36864};
  const int* bases[3]     = {base1, base2, base3};
  float*     Cbuf[3]      = {C1, C2, C3};

  float* outp = (float*)d_out;

  // bucket nodes by type (re-done every call: ws is not re-initialized)
  reset_counts<<<dim3(1), dim3(32), 0, stream>>>(counts);
  build_type_lists<<<dim3((B_N + 255) / 256), dim3(256), 0, stream>>>(
      nt, counts, lists, B_N);

  const int apply_blocks = (B_N + GNODES - 1) / GNODES;

  for (int e = 0; e < NTYPE; ++e) {
    for (int l = 0; l < 4; ++l) {
      const int mcount = (l == 0) ? 1 : (2 * l + 1);
      for (int nu = 0; nu < 3; ++nu) {
        // dict order: A, node_types, then per l: (U_1,W_1,U_2,W_2,U_3,W_3)
        const float* U = (const float*)d_in[2 + l * 6 + nu * 2];
        const float* W = (const float*)d_in[2 + l * 6 + nu * 2 + 1];
        const int R = mcount * rows_unit[nu];
        const int K = Kn[nu];
        float* C = Cbuf[nu] + (size_t)bases[nu][l] * FEAT;
        dim3 grid(R / 16, FEAT / 16);
        cg_weight_gemm_wmma<<<grid, dim3(32), 0, stream>>>(U, W, C, R, K, e);
      }
    }
    apply_contraction_grouped<<<dim3(apply_blocks), dim3(FEAT), 0, stream>>>(
        A, C3, C2, C1, counts, lists, outp, e, B_N);
  }
}